// TriAddAttention_86122684220185
// MI455X (gfx1250) — compile-verified
//
#include <hip/hip_runtime.h>
#include <math.h>

typedef __attribute__((ext_vector_type(2))) float v2f;
typedef __attribute__((ext_vector_type(8))) float v8f;
typedef __attribute__((ext_vector_type(4))) unsigned int u32x4;
typedef __attribute__((ext_vector_type(8))) unsigned int u32x8;

#define LOG2E 1.4426950408889634f

#define LA 128
#define LB 96
#define LC 12
#define DH 64
#define KCC (LB * LC) /* 1152 */
#define DMODEL 512

// Hardware transcendental tanh (gfx1250 V_TANH_F32). v_nop covers the
// TRANS-op co-execution hazard (ISA 5.7.4) since the compiler cannot see
// through the asm to schedule the dependency itself.
static __device__ __forceinline__ float fast_tanh(float x) {
  float y;
  asm("v_tanh_f32 %0, %1\n\tv_nop" : "=v"(y) : "v"(x));
  return y;
}

static __device__ __forceinline__ float wave_reduce_max(float v) {
#pragma unroll
  for (int m = 16; m > 0; m >>= 1) v = fmaxf(v, __shfl_xor(v, m, 32));
  return v;
}
static __device__ __forceinline__ float wave_reduce_sum(float v) {
#pragma unroll
  for (int m = 16; m > 0; m >>= 1) v += __shfl_xor(v, m, 32);
  return v;
}

// ---------------------------------------------------------------------------
// TDM: DMA a 2D tile [tile_y rows x tile_x f32] from global (row stride
// row_stride elements) into LDS at byte offset lds_off, packed contiguously.
// D# layout per CDNA5 ISA 8.3/8.4: group0 = {flags, lds_addr, global_addr,
// type=2}; group1 = {data_size=4B, tensor dims, tile dims, strides}.
// Issued via inline asm (portable across toolchains); TENSORcnt tracked.
// ---------------------------------------------------------------------------
static __device__ __forceinline__ void tdm_load_2d_f32(const float* gptr,
                                                       unsigned lds_off,
                                                       unsigned tile_x,
                                                       unsigned tile_y,
                                                       unsigned row_stride) {
  const unsigned long long ga = (unsigned long long)(uintptr_t)gptr;
  u32x4 g0;
  g0[0] = 1u;                                        /* count=1, user D# */
  g0[1] = lds_off;                                   /* lds_addr (bytes) */
  g0[2] = (unsigned)(ga & 0xFFFFFFFFu);              /* global_addr[31:0] */
  g0[3] = (unsigned)((ga >> 32) & 0x01FFFFFFu) | (2u << 30); /* [56:32]|type=2 */

  const unsigned tdim0 = row_stride; /* x extent for OOB (>= tile_x) */
  const unsigned tdim1 = tile_y;     /* y extent for OOB */
  u32x8 g1;
  g1[0] = (2u << 16);                                 /* data_size=4B */
  g1[1] = (tdim0 & 0xFFFFu) << 16;                    /* tensor_dim0 lo16 */
  g1[2] = (tdim0 >> 16) | ((tdim1 & 0xFFFFu) << 16);  /* dim0 hi | dim1 lo */
  g1[3] = (tdim1 >> 16) | ((tile_x & 0xFFFFu) << 16); /* dim1 hi | tile_dim0 */
  g1[4] = (tile_y & 0xFFFFu);                         /* tile_dim1; tile_dim2=0 */
  g1[5] = row_stride;                                 /* tensor_dim0_stride lo32 */
  g1[6] = 0u;                                         /* stride0 hi | stride1 lo */
  g1[7] = 0u;                                         /* stride1 hi */

  u32x4 g2 = {0u, 0u, 0u, 0u};
  u32x4 g3 = {0u, 0u, 0u, 0u};
  asm volatile("tensor_load_to_lds %0, %1, %2, %3"
               :
               : "s"(g0), "s"(g1), "s"(g2), "s"(g3)
               : "memory");
}

// ---------------------------------------------------------------------------
// Projection GEMM: Out[M,512] = A[M,512] @ W[512,512] + bias
// One wave32 per 16x16 output tile, V_WMMA_F32_16X16X4_F32, K-step 4.
// ---------------------------------------------------------------------------
__global__ __launch_bounds__(128) void proj_wmma_kernel(
    const float* __restrict__ A, const float* __restrict__ W,
    const float* __restrict__ bias, float* __restrict__ Out, int M) {
  const int K = DMODEL, N = DMODEL;
  const int lane = threadIdx.x & 31;
  const int wave = threadIdx.x >> 5;
  const int tilesN = N / 16; /* 32 */
  const int tilesM = (M + 15) >> 4;
  const int numTiles = tilesM * tilesN;
  const int tileId = blockIdx.x * 4 + wave;
  if (tileId >= numTiles) return;
  const int tm = tileId / tilesN;
  const int tn = tileId - tm * tilesN;

  const int half = lane >> 4;
  const int l16 = lane & 15;
  const int mrow = tm * 16 + l16;
  const int arow = (mrow < M) ? mrow : (M - 1);
  const int ncol = tn * 16 + l16;

  v8f acc = {};
  for (int k0 = 0; k0 < K; k0 += 4) {
    const int ka = k0 + 2 * half;
    v2f a;
    a.x = A[(size_t)arow * K + ka];
    a.y = A[(size_t)arow * K + ka + 1];
    v2f b;
    b.x = W[(size_t)ka * N + ncol];
    b.y = W[(size_t)(ka + 1) * N + ncol];
    acc = __builtin_amdgcn_wmma_f32_16x16x4_f32(false, a, false, b, (short)0,
                                                acc, false, false);
  }
  const float bn = bias[ncol];
#pragma unroll
  for (int r = 0; r < 8; ++r) {
    int row = tm * 16 + r + 8 * half;
    if (row < M) Out[(size_t)row * N + ncol] = acc[r] + bn;
  }
}

// ---------------------------------------------------------------------------
// Fused TriAdd attention per (b,h): TDM-staged LDS -> v_tanh scores ->
// online softmax -> attn to global (required output) -> WMMA attn @ (V*C).
// Block = 256 threads = 8 waves; wave w owns q rows [16w, 16w+16).
// ---------------------------------------------------------------------------
__global__ __launch_bounds__(256) void triadd_attn_kernel(
    const float* __restrict__ Qp, const float* __restrict__ Kp,
    const float* __restrict__ Vp, const float* __restrict__ Cp,
    const float* __restrict__ p, float* __restrict__ Out,
    float* __restrict__ Attn) {
  extern __shared__ float smem[];
  float* Ks = smem;              /* LB*DH */
  float* Vs = Ks + LB * DH;      /* LB*DH */
  float* Cs = Vs + LB * DH;      /* LC*DH */
  float* ps = Cs + LC * DH;      /* DH    */
  float* Qs = ps + DH;           /* LA*DH */
  float* stM = Qs + LA * DH;     /* 8*16  */
  float* stS = stM + 8 * 16;     /* 8*16  */

  const int bh = blockIdx.x; /* b*8 + h */
  const int b = bh >> 3, h = bh & 7;
  const int tid = threadIdx.x;
  const int lane = tid & 31, wave = tid >> 5;
  const int hoff = h * DH;

  /* ---- TDM staging: K, V, C, Q head slices DMA'd into LDS ------------- */
  const unsigned ldsBase = __builtin_amdgcn_groupstaticsize();
  if (wave == 0) {
    tdm_load_2d_f32(Kp + ((size_t)b * LB) * DMODEL + hoff,
                    ldsBase + 0u, DH, LB, DMODEL);
    tdm_load_2d_f32(Vp + ((size_t)b * LB) * DMODEL + hoff,
                    ldsBase + (unsigned)(LB * DH) * 4u, DH, LB, DMODEL);
    tdm_load_2d_f32(Cp + ((size_t)b * LC) * DMODEL + hoff,
                    ldsBase + (unsigned)(2 * LB * DH) * 4u, DH, LC, DMODEL);
    tdm_load_2d_f32(Qp + ((size_t)b * LA) * DMODEL + hoff,
                    ldsBase + (unsigned)(2 * LB * DH + LC * DH + DH) * 4u, DH,
                    LA, DMODEL);
    __builtin_amdgcn_s_wait_tensorcnt(0);
  }
  for (int i = tid; i < DH; i += 256) ps[i] = p[h * DH + i];
  __syncthreads();

  const int q0 = wave * 16;
  float* attnBase = Attn + ((size_t)bh * LA + q0) * KCC;

  /* -------- Phase A: raw scores + online softmax stats ------------------ */
  for (int qi = 0; qi < 16; ++qi) {
    const float* qrow = Qs + (q0 + qi) * DH;
    float* arow = attnBase + (size_t)qi * KCC;
    float Mx = -3.0e38f, S = 0.f;
    for (int kc0 = 0; kc0 < KCC; kc0 += 32) {
      const int kc = kc0 + lane;
      const int k = kc / LC, c = kc - k * LC;
      const float* krow = Ks + k * DH;
      const float* crow = Cs + c * DH;
      float s = 0.f;
#pragma unroll 8
      for (int d = 0; d < DH; ++d)
        s += ps[d] * fast_tanh(qrow[d] + krow[d] + crow[d]);
      arow[kc] = s; /* raw score; normalized in Phase B */
      const float Mn = fmaxf(Mx, wave_reduce_max(s));
      const float e = __builtin_amdgcn_exp2f((s - Mn) * LOG2E);
      S = S * __builtin_amdgcn_exp2f((Mx - Mn) * LOG2E) + wave_reduce_sum(e);
      Mx = Mn;
    }
    if (lane == 0) {
      stM[wave * 16 + qi] = Mx;
      stS[wave * 16 + qi] = S;
    }
  }

  /* -------- Phase B: normalize attn in place (required output) ---------- */
  for (int qi = 0; qi < 16; ++qi) {
    const float Mx = stM[wave * 16 + qi];
    const float invS = 1.f / stS[wave * 16 + qi];
    float* arow = attnBase + (size_t)qi * KCC;
    for (int kc = lane; kc < KCC; kc += 32)
      arow[kc] = __builtin_amdgcn_exp2f((arow[kc] - Mx) * LOG2E) * invS;
  }

  /* -------- Phase C: out[16,64] = attn[16,1152] @ VC[1152,64] ----------- */
  const int half = lane >> 4, l16 = lane & 15;
  v8f acc[4] = {v8f{}, v8f{}, v8f{}, v8f{}};
  const float* arowA = attnBase + (size_t)l16 * KCC; /* A: M = lane%16 */
  for (int kc0 = 0; kc0 < KCC; kc0 += 4) {
    const int kb = kc0 + 2 * half;
    __builtin_prefetch(arowA + kb + 64, 0, 3); /* global_prefetch_b8 */
    v2f a;
    a.x = arowA[kb];
    a.y = arowA[kb + 1];
    const int k0i = kb / LC, c0i = kb - k0i * LC;
    const int k1i = (kb + 1) / LC, c1i = (kb + 1) - k1i * LC;
    const float* v0 = Vs + k0i * DH;
    const float* c0 = Cs + c0i * DH;
    const float* v1 = Vs + k1i * DH;
    const float* c1 = Cs + c1i * DH;
#pragma unroll
    for (int nt = 0; nt < 4; ++nt) {
      const int n = nt * 16 + l16;
      v2f bb;
      bb.x = v0[n] * c0[n]; /* VC[kb  ][n] */
      bb.y = v1[n] * c1[n]; /* VC[kb+1][n] */
      acc[nt] = __builtin_amdgcn_wmma_f32_16x16x4_f32(false, a, false, bb,
                                                      (short)0, acc[nt], false,
                                                      false);
    }
  }
  float* obase = Out + ((size_t)b * LA) * DMODEL + hoff;
#pragma unroll
  for (int nt = 0; nt < 4; ++nt) {
#pragma unroll
    for (int r = 0; r < 8; ++r) {
      const int qrow = q0 + r + 8 * half; /* C/D: M = r + 8*(lane/16) */
      obase[(size_t)qrow * DMODEL + nt * 16 + l16] = acc[nt][r];
    }
  }
}

// ---------------------------------------------------------------------------
extern "C" void kernel_launch(void* const* d_in, const int* in_sizes, int n_in,
                              void* d_out, int out_size, void* d_ws,
                              size_t ws_size, hipStream_t stream) {
  (void)in_sizes; (void)n_in; (void)out_size; (void)ws_size;
  const float* Q_in = (const float*)d_in[0];
  const float* K_in = (const float*)d_in[1];
  const float* V_in = (const float*)d_in[2];
  const float* C_in = (const float*)d_in[3];
  const float* Wq = (const float*)d_in[4];
  const float* bq = (const float*)d_in[5];
  const float* Wk = (const float*)d_in[6];
  const float* bk = (const float*)d_in[7];
  const float* Wv = (const float*)d_in[8];
  const float* bv = (const float*)d_in[9];
  const float* Wc = (const float*)d_in[10];
  const float* bc = (const float*)d_in[11];
  const float* p = (const float*)d_in[12];

  float* ws = (float*)d_ws;
  float* Qp = ws;                      /* [2*128,512] */
  float* Kp = Qp + 2 * LA * DMODEL;    /* [2*96,512]  */
  float* Vp = Kp + 2 * LB * DMODEL;    /* [2*96,512]  */
  float* Cp = Vp + 2 * LB * DMODEL;    /* [2*12,512]  */

  float* Out = (float*)d_out;              /* [2,128,512] */
  float* Attn = Out + 2 * LA * DMODEL;     /* [2,8,128,1152] */

  auto launch_proj = [&](const float* A, const float* W, const float* bias,
                         float* O, int M) {
    const int tilesM = (M + 15) / 16;
    const int numTiles = tilesM * (DMODEL / 16);
    const int blocks = (numTiles + 3) / 4;
    hipLaunchKernelGGL(proj_wmma_kernel, dim3(blocks), dim3(128), 0, stream, A,
                       W, bias, O, M);
  };
  launch_proj(Q_in, Wq, bq, Qp, 2 * LA);
  launch_proj(K_in, Wk, bk, Kp, 2 * LB);
  launch_proj(V_in, Wv, bv, Vp, 2 * LB);
  launch_proj(C_in, Wc, bc, Cp, 2 * LC);

  const size_t smemBytes =
      (size_t)(2 * LB * DH + LC * DH + DH + LA * DH + 2 * 8 * 16) *
      sizeof(float);
  hipLaunchKernelGGL(triadd_attn_kernel, dim3(16), dim3(256), smemBytes,
                     stream, Qp, Kp, Vp, Cp, p, Out, Attn);
}